// RelationAwareGraphModel_10857677325088
// MI455X (gfx1250) — compile-verified
//
#include <hip/hip_runtime.h>
#include <hip/hip_bf16.h>

typedef __attribute__((ext_vector_type(2))) float v2f;
typedef __attribute__((ext_vector_type(8))) float v8f;

#define H 128
#define R 8
#define KTOT 1152          // 128 (self) + 1024 (rel)
#define NCHUNK 9           // 1152 / 128

// ---------------------------------------------------------------------------
// Pack B for both layers in WMMA fragment-pair order:
//   logical B[k][d]:  k < 128 -> self_W[l][d][k]
//                     k >= 128 -> rel_W[l][r][d][h], r=(k-128)/128, h=(k-128)%128
// Stored as float2 pairs: Bm[l][(k>>1)*256 + d*2 + (k&1)]
// so a lane's B fragment {B[k][d], B[k+1][d]} is one contiguous b64 load.
// ---------------------------------------------------------------------------
__global__ void packB_kernel(const float* __restrict__ selfW,
                             const float* __restrict__ relW,
                             float* __restrict__ Bmat) {
    int idx = blockIdx.x * blockDim.x + threadIdx.x;
    if (idx >= 2 * KTOT * H) return;
    int d = idx & (H - 1);
    int k = (idx >> 7) % KTOT;
    int l = idx / (KTOT * H);
    float v;
    if (k < H) {
        v = selfW[((size_t)l * H + d) * H + k];
    } else {
        int kk = k - H;
        int r = kk >> 7;
        int h = kk & (H - 1);
        v = relW[(((size_t)l * R + r) * H + d) * H + h];
    }
    Bmat[(size_t)l * KTOT * H + (size_t)(k >> 1) * (2 * H) + d * 2 + (k & 1)] = v;
}

// ---------------------------------------------------------------------------
// x[n][h] = node_emb[type[n]][h]
// ---------------------------------------------------------------------------
__global__ void gather_kernel(const int* __restrict__ types,
                              const float* __restrict__ emb,
                              float* __restrict__ x, int n) {
    int idx = blockIdx.x * blockDim.x + threadIdx.x;
    if (idx >= n * H) return;
    int node = idx >> 7;
    int h = idx & (H - 1);
    x[idx] = emb[(size_t)types[node] * H + h];
}

// ---------------------------------------------------------------------------
// Edge scatter: agg[(tgt*R + rel)][:] += x[src][:].  One wave per edge,
// each lane moves 4 consecutive floats (128/32). x is L2-resident (51 MB).
// ---------------------------------------------------------------------------
__global__ __launch_bounds__(256) void scatter_kernel(
    const float* __restrict__ x, const int* __restrict__ ei,
    const int* __restrict__ et, float* __restrict__ agg, int nedges) {
    int wid = blockIdx.x * 8 + (threadIdx.x >> 5);
    if (wid >= nedges) return;
    int lane = threadIdx.x & 31;
    int2 e = ((const int2*)ei)[wid];      // (src, tgt)
    int rel = et[wid];
    float4 v = *(const float4*)(x + (size_t)e.x * H + lane * 4);
    float* dst = agg + ((size_t)e.y * R + rel) * H + lane * 4;
    atomicAdd(dst + 0, v.x);
    atomicAdd(dst + 1, v.y);
    atomicAdd(dst + 2, v.z);
    atomicAdd(dst + 3, v.w);
}

// ---------------------------------------------------------------------------
// Fused layer GEMM: x_out = relu( [x | agg] (N x 1152) @ B (1152 x 128) + b )
// Block = 256 threads (8 waves) computes 64 rows x 128 cols.
//   wave w -> column tile [w*16, w*16+16), 4 row tiles (rg = 0..3).
//   One B fragment (single b64 load from pair-packed B) feeds 4 WMMAs
//   against 4 A fragments from LDS -> B read exactly once per block.
// A chunk (64 x 128) staged in LDS with 132-float pitch (bank-conflict free).
// Last block clamps A-source rows and guards stores (N % 64 != 0 is OK).
// ---------------------------------------------------------------------------
__global__ __launch_bounds__(256) void gemm_layer_kernel(
    const float* __restrict__ x_in,   // N x 128
    const float* __restrict__ agg,    // N x 1024
    const float* __restrict__ Bmat,   // this layer, pair-packed (1152 x 128)
    const float* __restrict__ bias,   // 128
    float* __restrict__ x_out,        // N x 128
    int n) {
    __shared__ float As[64][132];

    int t = threadIdx.x;
    int wave = t >> 5, lane = t & 31;
    int laneHalf = lane >> 4;          // 0: K={0,1}, 1: K={2,3}
    int lane16 = lane & 15;
    int m0 = blockIdx.x * 64;
    int col = wave * 16 + lane16;

    v8f c[4];
#pragma unroll
    for (int rg = 0; rg < 4; ++rg) c[rg] = (v8f){};

    for (int ch = 0; ch < NCHUNK; ++ch) {
        // ---- stage A chunk: rows m0..m0+63, k = ch*128 .. ch*128+127 ----
        {
            int r = t >> 3;            // 0..31
            int c4 = t & 7;            // 8 threads per row
#pragma unroll
            for (int half = 0; half < 2; ++half) {
                int row = r + half * 32;
                int grow = m0 + row;
                if (grow >= n) grow = n - 1;   // clamp: stay in-bounds
                const float* srcRow = (ch == 0)
                    ? (x_in + (size_t)grow * H)
                    : (agg + (size_t)grow * (R * H) + (ch - 1) * H);
#pragma unroll
                for (int j = 0; j < 4; ++j) {
                    int cc = (c4 + j * 8) * 4;
                    *(float4*)&As[row][cc] = *(const float4*)(srcRow + cc);
                }
            }
        }
        __syncthreads();

        const float2* Bp = (const float2*)(Bmat + (size_t)ch * H * H);
#pragma unroll
        for (int ks = 0; ks < 128; ks += 4) {
            // B fragment: one b64 load of {B[kb][col], B[kb+1][col]}
            int pp = (ks >> 1) + laneHalf;         // pair index in chunk
            float2 bf = Bp[(size_t)pp * H + col];
            v2f b; b.x = bf.x; b.y = bf.y;
            // 4 A fragments from LDS share this B fragment
#pragma unroll
            for (int rg = 0; rg < 4; ++rg) {
                float2 af = *(const float2*)&As[rg * 16 + lane16][ks + laneHalf * 2];
                v2f a; a.x = af.x; a.y = af.y;
                c[rg] = __builtin_amdgcn_wmma_f32_16x16x4_f32(
                    false, a, false, b, (short)0, c[rg], false, false);
            }
        }
        __syncthreads();
    }

    // ---- epilogue: bias + relu; C/D layout: VGPR i -> rows i and i+8 ----
    float bv = bias[col];
#pragma unroll
    for (int rg = 0; rg < 4; ++rg) {
#pragma unroll
        for (int i = 0; i < 8; ++i) {
            int m = m0 + rg * 16 + i + laneHalf * 8;
            if (m < n) {
                float v = fmaxf(c[rg][i] + bv, 0.0f);
                x_out[(size_t)m * H + col] = v;
            }
        }
    }
}

// ---------------------------------------------------------------------------
// Column-wise sum + max over final x (post-ReLU => x >= 0, so uint max and
// 0-init are valid).  Grid-strided rows, coalesced 512B row reads.
// ---------------------------------------------------------------------------
__global__ __launch_bounds__(128) void reduce_kernel(
    const float* __restrict__ x, float* __restrict__ gsum,
    unsigned* __restrict__ gmaxu, int n) {
    int col = threadIdx.x;
    float s = 0.0f, m = 0.0f;
    for (int row = blockIdx.x; row < n; row += gridDim.x) {
        float v = x[(size_t)row * H + col];
        s += v;
        m = fmaxf(m, v);
    }
    atomicAdd(&gsum[col], s);
    atomicMax(&gmaxu[col], __float_as_uint(m));
}

// ---------------------------------------------------------------------------
// Head: feat MLP + comb + bin/mc heads -> 7 output floats. Single block.
// ---------------------------------------------------------------------------
__global__ __launch_bounds__(256) void head_kernel(
    const float* __restrict__ x, const float* __restrict__ gsum,
    const unsigned* __restrict__ gmaxu, const int* __restrict__ anchor_p,
    const float* __restrict__ numeric, const float* __restrict__ boolean_,
    const int* __restrict__ cats, const float* __restrict__ ce0,
    const float* __restrict__ ce1, const float* __restrict__ ce2,
    const float* __restrict__ fW, const float* __restrict__ fb,
    const float* __restrict__ bW1, const float* __restrict__ bb1,
    const float* __restrict__ bW2, const float* __restrict__ bb2,
    const float* __restrict__ mW1, const float* __restrict__ mb1,
    const float* __restrict__ mW2, const float* __restrict__ mb2,
    float* __restrict__ out, int n) {
    __shared__ float feat[72];
    __shared__ float comb[432];
    __shared__ float hbin[192];
    __shared__ float hmc[64];
    int t = threadIdx.x;

    if (t < 72) {
        float v;
        if (t < 32)      v = numeric[t];
        else if (t < 48) v = boolean_[t - 32];
        else if (t < 56) v = ce0[cats[0] * 8 + (t - 48)];
        else if (t < 64) v = ce1[cats[1] * 8 + (t - 56)];
        else             v = ce2[cats[2] * 8 + (t - 64)];
        feat[t] = v;
    }
    __syncthreads();

    int anchor = anchor_p[0];
    float invN = 1.0f / (float)n;
    if (t < 128) {
        comb[t]       = x[(size_t)anchor * H + t];
        comb[128 + t] = gsum[t] * invN;
        comb[256 + t] = __uint_as_float(gmaxu[t]);
    } else if (t < 176) {
        int d = t - 128;                       // enc: 48 outputs, 72 inputs
        float acc = fb[d];
        for (int i = 0; i < 72; ++i) acc += fW[d * 72 + i] * feat[i];
        comb[384 + d] = fmaxf(acc, 0.0f);
    }
    __syncthreads();

    if (t < 192) {                             // 3 bin heads x 64 hidden
        int k = t >> 6, hh = t & 63;
        const float* wr = bW1 + (size_t)(k * 64 + hh) * 432;
        float acc = bb1[k * 64 + hh];
        for (int i = 0; i < 432; ++i) acc += wr[i] * comb[i];
        hbin[t] = fmaxf(acc, 0.0f);
    } else {                                   // mc head: 64 hidden
        int hh = t - 192;
        const float* wr = mW1 + (size_t)hh * 432;
        float acc = mb1[hh];
        for (int i = 0; i < 432; ++i) acc += wr[i] * comb[i];
        hmc[hh] = fmaxf(acc, 0.0f);
    }
    __syncthreads();

    if (t < 3) {
        float acc = bb2[t];
        for (int i = 0; i < 64; ++i) acc += bW2[t * 64 + i] * hbin[t * 64 + i];
        out[t] = acc;
    } else if (t >= 8 && t < 12) {
        int s = t - 8;
        float acc = mb2[s];
        for (int i = 0; i < 64; ++i) acc += mW2[s * 64 + i] * hmc[i];
        out[3 + s] = acc;
    }
}

// ---------------------------------------------------------------------------
extern "C" void kernel_launch(void* const* d_in, const int* in_sizes, int n_in,
                              void* d_out, int out_size, void* d_ws, size_t ws_size,
                              hipStream_t stream) {
    const int*   node_types = (const int*)d_in[0];
    const int*   edge_index = (const int*)d_in[1];
    const int*   edge_types = (const int*)d_in[2];
    const int*   categorical= (const int*)d_in[3];
    const int*   anchor     = (const int*)d_in[4];
    const float* numeric    = (const float*)d_in[5];
    const float* boolean_   = (const float*)d_in[6];
    const float* node_emb   = (const float*)d_in[7];
    const float* self_W     = (const float*)d_in[8];
    const float* self_b     = (const float*)d_in[9];
    const float* rel_W      = (const float*)d_in[10];
    const float* cat0       = (const float*)d_in[11];
    const float* cat1       = (const float*)d_in[12];
    const float* cat2       = (const float*)d_in[13];
    const float* feat_W     = (const float*)d_in[14];
    const float* feat_b     = (const float*)d_in[15];
    const float* bin_W1     = (const float*)d_in[16];
    const float* bin_b1     = (const float*)d_in[17];
    const float* bin_W2     = (const float*)d_in[18];
    const float* bin_b2     = (const float*)d_in[19];
    const float* mc_W1      = (const float*)d_in[20];
    const float* mc_b1      = (const float*)d_in[21];
    const float* mc_W2      = (const float*)d_in[22];
    const float* mc_b2      = (const float*)d_in[23];

    const int N = in_sizes[0];
    const int E = in_sizes[2];

    // workspace layout (fp32)
    float* ws   = (float*)d_ws;
    float* x0   = ws;                                // N*128
    float* x1   = x0 + (size_t)N * H;                // N*128
    float* agg  = x1 + (size_t)N * H;                // N*1024
    float* Bm   = agg + (size_t)N * R * H;           // 2*1152*128 (pair-packed)
    float* gsum = Bm + (size_t)2 * KTOT * H;         // 128
    unsigned* gmaxu = (unsigned*)(gsum + H);         // 128

    // pack B (both layers) + gather node embeddings
    packB_kernel<<<(2 * KTOT * H + 255) / 256, 256, 0, stream>>>(self_W, rel_W, Bm);
    gather_kernel<<<((size_t)N * H + 255) / 256, 256, 0, stream>>>(node_types, node_emb, x0, N);

    float* xin = x0;
    float* xout = x1;
    for (int l = 0; l < 2; ++l) {
        hipMemsetAsync(agg, 0, (size_t)N * R * H * sizeof(float), stream);
        scatter_kernel<<<(E + 7) / 8, 256, 0, stream>>>(xin, edge_index, edge_types, agg, E);
        gemm_layer_kernel<<<(N + 63) / 64, 256, 0, stream>>>(
            xin, agg, Bm + (size_t)l * KTOT * H, self_b + (size_t)l * H, xout, N);
        float* tmp = xin; xin = xout; xout = tmp;
    }
    // final x is in `xin` (== x0 after 2 layers)

    hipMemsetAsync(gsum, 0, 2 * H * sizeof(float), stream);  // gsum + gmax
    reduce_kernel<<<512, 128, 0, stream>>>(xin, gsum, gmaxu, N);

    head_kernel<<<1, 256, 0, stream>>>(
        xin, gsum, gmaxu, anchor, numeric, boolean_, categorical,
        cat0, cat1, cat2, feat_W, feat_b, bin_W1, bin_b1, bin_W2, bin_b2,
        mc_W1, mc_b1, mc_W2, mc_b2, (float*)d_out, N);
}